// CANLayer_82695300317534
// MI455X (gfx1250) — compile-verified
//
#include <hip/hip_runtime.h>
#include <hip/hip_bf16.h>

// CAN layer for MI455X (gfx1250, wave32).
//   N=50000 nodes, IN_CH=128, HEADS=4, OUT_CH=32 (HC=128), E=400000 edges/path.
// GEMMs use V_WMMA_F32_16X16X4_F32 (full f32 precision, matches f32 reference).
// Edge softmax + aggregation use wave-per-edge gathers and global f32 atomics;
// working set (~90MB) fits the 192MB L2, so scatter/gather is L2-resident.
//
// Workspace layout (floats): xm[N*128] | e_buf[E*4] | denom[N*4]  (~33 MB),
// reused for the lower and upper paths (stream-serialized kernels).

#define IN_CH   128
#define HC      128
#define HEADS   4
#define OUT_CH  32
#define EPS_SKIP 1.000001f

typedef float v2f __attribute__((ext_vector_type(2)));
typedef float v8f __attribute__((ext_vector_type(8)));

// ---------------------------------------------------------------------------
// out[rt*16 .. rt*16+15][ct*16 .. ct*16+15] = scale * X[16x128] @ W[128x16]
// One wave per 16x16 tile. K = 128 in steps of 4 (32 WMMA ops).
// A layout (32-bit 16x4): lanes 0-15 rows M=0..15, VGPR0/1 = K {0,1}; lanes
// 16-31 same rows, K {2,3}. B layout (4x16): lanes 0-15 cols, rows K {0,1};
// lanes 16-31 cols, rows K {2,3}. C/D: VGPR v -> M = v + 8*(lane>=16), N=lane&15.
// ---------------------------------------------------------------------------
__global__ void __launch_bounds__(32)
wmma_gemm_f32(const float* __restrict__ X, const float* __restrict__ W,
              float* __restrict__ out, float scale)
{
    const int rt   = blockIdx.x;      // row tile (N/16 tiles)
    const int ct   = blockIdx.y;      // col tile (HC/16 = 8 tiles)
    const int lane = threadIdx.x;     // 0..31
    const int half = lane >> 4;       // 0 or 1
    const int r    = lane & 15;

    const float* Xrow = X + (size_t)(rt * 16 + r) * IN_CH;
    const float* Wcol = W + ct * 16 + r;

    v8f acc = {};
#pragma unroll 4
    for (int k0 = 0; k0 < IN_CH; k0 += 4) {
        const int ka = k0 + 2 * half;
        v2f a, b;
        a.x = Xrow[ka];
        a.y = Xrow[ka + 1];
        b.x = Wcol[(size_t)ka * HC];
        b.y = Wcol[(size_t)(ka + 1) * HC];
        acc = __builtin_amdgcn_wmma_f32_16x16x4_f32(
            /*neg_a=*/false, a, /*neg_b=*/false, b,
            /*c_mod=*/(short)0, acc, /*reuse_a=*/false, /*reuse_b=*/false);
    }

    float* orow = out + (size_t)(rt * 16 + 8 * half) * HC + ct * 16 + r;
#pragma unroll
    for (int v = 0; v < 8; ++v)
        orow[(size_t)v * HC] = acc[v] * scale;
}

// ---------------------------------------------------------------------------
// Pass 1: one wave per edge. Per-head attention logit via 32-lane dot,
// leaky_relu(0.01), exp -> e_buf[e][h]; atomicAdd into denom[tgt][h].
// ---------------------------------------------------------------------------
__global__ void __launch_bounds__(256)
attn_pass1(const float* __restrict__ xm, const float* __restrict__ a_src,
           const float* __restrict__ a_dst, const int* __restrict__ tgt,
           const int* __restrict__ src, float* __restrict__ e_buf,
           float* __restrict__ denom, int E)
{
    const int e    = (int)((blockIdx.x * (unsigned)blockDim.x + threadIdx.x) >> 5);
    const int lane = threadIdx.x & 31;
    if (e >= E) return;

    const int s = src[e];
    const int t = tgt[e];
    const float* xs = xm + (size_t)s * HC;
    const float* xt = xm + (size_t)t * HC;

    float p[HEADS];
#pragma unroll
    for (int h = 0; h < HEADS; ++h) {
        const int c = h * OUT_CH + lane;
        p[h] = xs[c] * a_src[c] + xt[c] * a_dst[c];
    }
    // wave32 butterfly reduction, all 4 heads in flight
#pragma unroll
    for (int off = 16; off; off >>= 1) {
#pragma unroll
        for (int h = 0; h < HEADS; ++h)
            p[h] += __shfl_xor(p[h], off, 32);
    }
    if (lane < HEADS) {
        float a = p[lane];
        a = a > 0.0f ? a : 0.01f * a;      // leaky_relu
        const float ev = expf(a);
        e_buf[(size_t)e * HEADS + lane] = ev;
        atomicAdd(&denom[(size_t)t * HEADS + lane], ev);
    }
}

// ---------------------------------------------------------------------------
// Pass 2: one wave per edge. alpha = e/(denom[tgt]+1e-16); scatter-add
// alpha-weighted source features into out[tgt][:].
// ---------------------------------------------------------------------------
__global__ void __launch_bounds__(256)
attn_pass2(const float* __restrict__ xm, const float* __restrict__ e_buf,
           const float* __restrict__ denom, const int* __restrict__ tgt,
           const int* __restrict__ src, float* __restrict__ out, int E)
{
    const int e    = (int)((blockIdx.x * (unsigned)blockDim.x + threadIdx.x) >> 5);
    const int lane = threadIdx.x & 31;
    if (e >= E) return;

    const int s = src[e];
    const int t = tgt[e];

    float v = 0.0f;
    if (lane < HEADS)
        v = e_buf[(size_t)e * HEADS + lane] /
            (denom[(size_t)t * HEADS + lane] + 1e-16f);
    float alpha[HEADS];
#pragma unroll
    for (int h = 0; h < HEADS; ++h)
        alpha[h] = __shfl(v, h, 32);

    const float* xs = xm + (size_t)s * HC;
    float* od = out + (size_t)t * HC;
#pragma unroll
    for (int h = 0; h < HEADS; ++h) {
        const int c = h * OUT_CH + lane;
        atomicAdd(&od[c], xs[c] * alpha[h]);
    }
}

__global__ void __launch_bounds__(256)
zero_f32(float* __restrict__ p, int n)
{
    const int i = (int)(blockIdx.x * (unsigned)blockDim.x + threadIdx.x);
    if (i < n) p[i] = 0.0f;
}

__global__ void __launch_bounds__(256)
relu_inplace(float* __restrict__ p, int n)
{
    const int i = (int)(blockIdx.x * (unsigned)blockDim.x + threadIdx.x);
    if (i < n) p[i] = fmaxf(p[i], 0.0f);
}

extern "C" void kernel_launch(void* const* d_in, const int* in_sizes, int n_in,
                              void* d_out, int out_size, void* d_ws, size_t ws_size,
                              hipStream_t stream)
{
    const float* x          = (const float*)d_in[0];
    const float* W_low      = (const float*)d_in[1];
    const float* a_src_low  = (const float*)d_in[2];
    const float* a_dst_low  = (const float*)d_in[3];
    const float* W_up       = (const float*)d_in[4];
    const float* a_src_up   = (const float*)d_in[5];
    const float* a_dst_up   = (const float*)d_in[6];
    const float* W_skip     = (const float*)d_in[7];
    const int*   lower_tgt  = (const int*)d_in[8];
    const int*   lower_src  = (const int*)d_in[9];
    const int*   upper_tgt  = (const int*)d_in[10];
    const int*   upper_src  = (const int*)d_in[11];

    const int N = in_sizes[0] / IN_CH;   // 50000 (multiple of 16)
    const int E = in_sizes[8];           // 400000

    float* out   = (float*)d_out;
    float* xm    = (float*)d_ws;                    // N*HC
    float* e_buf = xm + (size_t)N * HC;             // E*HEADS
    float* denom = e_buf + (size_t)E * HEADS;       // N*HEADS

    const dim3 ggrid(N / 16, HC / 16);              // 3125 x 8 tiles
    const int  edge_blocks  = (E + 7) / 8;          // 8 waves per 256-thread block
    const int  denom_blocks = (N * HEADS + 255) / 256;
    const int  out_blocks   = (N * HC + 255) / 256;

    // Skip connection: out = (x @ W_skip) * EPS_SKIP  (accumulation base)
    wmma_gemm_f32<<<ggrid, 32, 0, stream>>>(x, W_skip, out, EPS_SKIP);

    // ---- lower path ----
    zero_f32<<<denom_blocks, 256, 0, stream>>>(denom, N * HEADS);
    wmma_gemm_f32<<<ggrid, 32, 0, stream>>>(x, W_low, xm, 1.0f);
    attn_pass1<<<edge_blocks, 256, 0, stream>>>(xm, a_src_low, a_dst_low,
                                                lower_tgt, lower_src,
                                                e_buf, denom, E);
    attn_pass2<<<edge_blocks, 256, 0, stream>>>(xm, e_buf, denom,
                                                lower_tgt, lower_src, out, E);

    // ---- upper path (reuses workspace; kernels serialize on the stream) ----
    zero_f32<<<denom_blocks, 256, 0, stream>>>(denom, N * HEADS);
    wmma_gemm_f32<<<ggrid, 32, 0, stream>>>(x, W_up, xm, 1.0f);
    attn_pass1<<<edge_blocks, 256, 0, stream>>>(xm, a_src_up, a_dst_up,
                                                upper_tgt, upper_src,
                                                e_buf, denom, E);
    attn_pass2<<<edge_blocks, 256, 0, stream>>>(xm, e_buf, denom,
                                                upper_tgt, upper_src, out, E);

    // Final fused update: relu in place
    relu_inplace<<<out_blocks, 256, 0, stream>>>(out, N * HC);
}